// PC_DeQuantizationModule_66494683676998
// MI455X (gfx1250) — compile-verified
//
#include <hip/hip_runtime.h>

#define NVOX 262144
#define KOFF 27
#define CH 16
#define NLAYER 6
#define NEG_SLOPE 0.2f

typedef __attribute__((ext_vector_type(2))) float v2f;
typedef __attribute__((ext_vector_type(4))) float v4f;
typedef __attribute__((ext_vector_type(8))) float v8f;

// ---------------------------------------------------------------------------
// conv1: Cin=1 -> Cout=16.  Pure VALU.  Writes both conv1 skip buffer and x.
// ---------------------------------------------------------------------------
__global__ __launch_bounds__(256) void conv_in_kernel(
    const float* __restrict__ feats, const int* __restrict__ idx,
    const unsigned char* __restrict__ mask, const float* __restrict__ W1,
    const float* __restrict__ b1, float* __restrict__ c1out,
    float* __restrict__ xout)
{
    __shared__ float sW[KOFF * CH];
    int tid = threadIdx.x;
    for (int t = tid; t < KOFF * CH; t += 256) sW[t] = W1[t];
    __syncthreads();

    int n = blockIdx.x * 256 + tid;
    float acc[CH];
#pragma unroll
    for (int c = 0; c < CH; ++c) acc[c] = b1[c];

    const int* ip = idx + (size_t)n * KOFF;
    const unsigned char* mp = mask + (size_t)n * KOFF;
    for (int k = 0; k < KOFF; ++k) {
        unsigned off = (unsigned)ip[k] * 4u;            // 32-bit byte offset
        float s = (float)mp[k];                         // u8 0/1 -> cvt
        float v = s * *(const float*)((const char*)feats + off);
#pragma unroll
        for (int c = 0; c < CH; ++c) acc[c] = fmaf(v, sW[k * CH + c], acc[c]);
    }
    float* o1 = c1out + (size_t)n * CH;
    float* ox = xout + (size_t)n * CH;
#pragma unroll
    for (int c = 0; c < CH; ++c) { o1[c] = acc[c]; ox[c] = acc[c]; }
}

// ---------------------------------------------------------------------------
// 16->16 sparse conv via V_WMMA_F32_16X16X4_F32, 2 tiles per wave.
//
// Chunk mapping: sub-GEMM kc covers channels {2kc, 2kc+1, 2kc+8, 2kc+9} with
// K = {0,1,2,3} -> channel {2kc, 2kc+1, 2kc+8, 2kc+9}.  Lane (m, hf) supplies
// K = 2hf,2hf+1, i.e. channels 8hf..8hf+7 across all four chunks — exactly
// TWO contiguous float4 (global_load_b128) gathers per tile per k.
// B fragment for chunk kc: W rows 2kc+8hf, 2kc+8hf+1 (ds_load pair, 64B gap).
//
// Depth-1 software pipeline; mask applied multiplicatively on loaded v4f
// (branch-free, EXEC all-ones at every WMMA); 32-bit gather offsets.
// ---------------------------------------------------------------------------
__global__ __launch_bounds__(256) void conv16_wmma_kernel(
    const float* __restrict__ xin, const int* __restrict__ idx,
    const unsigned char* __restrict__ mask, const float* __restrict__ W,
    const float* __restrict__ bias, const float* __restrict__ add,
    float* __restrict__ out, int relu)
{
    __shared__ float sW[KOFF * CH * CH];   // [k][ci][co]
    int tid = threadIdx.x;
    for (int t = tid; t < KOFF * CH * CH; t += 256) sW[t] = W[t];
    __syncthreads();

    int lane = tid & 31;
    int wv   = tid >> 5;
    int tile0 = (blockIdx.x * 8 + wv) * 2;     // this wave: tiles tile0, tile0+1
    int col  = lane & 15;                      // A row M ; B/D column N
    int hf   = lane >> 4;                      // half-wave sub-block
    int row0 = tile0 * 16 + col;
    int row1 = row0 + 16;
    float bn = bias[col];

    const int* ip0 = idx + (size_t)row0 * KOFF;
    const int* ip1 = idx + (size_t)row1 * KOFF;
    const unsigned char* mp0 = mask + (size_t)row0 * KOFF;
    const unsigned char* mp1 = mask + (size_t)row1 * KOFF;

    const char* xb = (const char*)xin;
    unsigned hoff = (unsigned)hf * 32u;        // lane's channel-range start (bytes)

    v8f acc0 = {}, acc1 = {};

    // ---- pipeline stage registers (current k): 2x float4 per tile ----
    v4f lo0, hi0, lo1, hi1;
    float s0, s1;
    {
        unsigned o0 = (unsigned)ip0[0] * 64u + hoff;
        unsigned o1 = (unsigned)ip1[0] * 64u + hoff;
        s0 = (float)mp0[0];
        s1 = (float)mp1[0];
        lo0 = *(const v4f*)(xb + o0);  hi0 = *(const v4f*)(xb + o0 + 16);
        lo1 = *(const v4f*)(xb + o1);  hi1 = *(const v4f*)(xb + o1 + 16);
    }

    for (int k = 0; k < KOFF; ++k) {
        // ---- prefetch next offset's fragments (in flight during WMMAs) ----
        int kn = (k + 1 < KOFF) ? (k + 1) : k;
        unsigned o0 = (unsigned)ip0[kn] * 64u + hoff;
        unsigned o1 = (unsigned)ip1[kn] * 64u + hoff;
        float s0n = (float)mp0[kn];
        float s1n = (float)mp1[kn];
        v4f lo0n = *(const v4f*)(xb + o0), hi0n = *(const v4f*)(xb + o0 + 16);
        v4f lo1n = *(const v4f*)(xb + o1), hi1n = *(const v4f*)(xb + o1 + 16);

        // ---- apply mask scale (v_pk_mul), spread into per-chunk fragments ----
        v4f m0l = lo0 * s0, m0h = hi0 * s0;
        v4f m1l = lo1 * s1, m1h = hi1 * s1;
        float f0[8] = {m0l.x, m0l.y, m0l.z, m0l.w, m0h.x, m0h.y, m0h.z, m0h.w};
        float f1[8] = {m1l.x, m1l.y, m1l.z, m1l.w, m1h.x, m1h.y, m1h.z, m1h.w};

        // ---- B fragments from LDS; each feeds both tiles ----
        const float* wk = &sW[k * CH * CH + hf * 8 * CH + col];
#pragma unroll
        for (int kc = 0; kc < 4; ++kc) {
            v2f b;
            b.x = wk[(kc * 2 + 0) * CH];
            b.y = wk[(kc * 2 + 1) * CH];
            v2f a0; a0.x = f0[kc * 2]; a0.y = f0[kc * 2 + 1];
            v2f a1; a1.x = f1[kc * 2]; a1.y = f1[kc * 2 + 1];
            acc0 = __builtin_amdgcn_wmma_f32_16x16x4_f32(
                false, a0, false, b, (short)0, acc0, false, false);
            acc1 = __builtin_amdgcn_wmma_f32_16x16x4_f32(
                false, a1, false, b, (short)0, acc1, false, false);
        }

        // ---- rotate pipeline ----
        lo0 = lo0n; hi0 = hi0n; lo1 = lo1n; hi1 = hi1n;
        s0 = s0n; s1 = s1n;
    }

    // D layout: lane holds column `col`; VGPR v holds row M = hf*8 + v
#pragma unroll
    for (int v = 0; v < 8; ++v) {
        int M = hf * 8 + v;
        size_t o0 = ((size_t)tile0 * 16 + M) * CH + col;
        size_t o1 = ((size_t)(tile0 + 1) * 16 + M) * CH + col;
        float v0 = acc0[v] + bn;
        float v1 = acc1[v] + bn;
        if (add) { v0 += add[o0]; v1 += add[o1]; }
        if (relu) {
            v0 = v0 > 0.0f ? v0 : NEG_SLOPE * v0;
            v1 = v1 > 0.0f ? v1 : NEG_SLOPE * v1;
        }
        out[o0] = v0;
        out[o1] = v1;
    }
}

// ---------------------------------------------------------------------------
// conv3: Cin=16 -> Cout=1 dot reduction + feats skip.  Mask multiplicative.
// ---------------------------------------------------------------------------
__global__ __launch_bounds__(256) void conv_out_kernel(
    const float* __restrict__ c2, const int* __restrict__ idx,
    const unsigned char* __restrict__ mask, const float* __restrict__ W3,
    const float* __restrict__ b3, const float* __restrict__ feats,
    float* __restrict__ out)
{
    __shared__ float sW[KOFF * CH];
    int tid = threadIdx.x;
    for (int t = tid; t < KOFF * CH; t += 256) sW[t] = W3[t];
    __syncthreads();

    int n = blockIdx.x * 256 + tid;
    float a = b3[0] + feats[n];
    const int* ip = idx + (size_t)n * KOFF;
    const unsigned char* mp = mask + (size_t)n * KOFF;
    const char* cb = (const char*)c2;
    for (int k = 0; k < KOFF; ++k) {
        unsigned off = (unsigned)ip[k] * 64u;           // 32-bit byte offset
        float s = (float)mp[k];
        const float* p = (const float*)(cb + off);      // unconditional gather
        float d = 0.0f;
#pragma unroll
        for (int c = 0; c < CH; ++c) d = fmaf(p[c], sW[k * CH + c], d);
        a = fmaf(s, d, a);
    }
    out[n] = a;
}

// ---------------------------------------------------------------------------
// Launch: 15 dependent conv passes; buffers in d_ws (48 MB, L2-resident).
// ---------------------------------------------------------------------------
extern "C" void kernel_launch(void* const* d_in, const int* in_sizes, int n_in,
                              void* d_out, int out_size, void* d_ws, size_t ws_size,
                              hipStream_t stream)
{
    const float* feats = (const float*)d_in[0];
    const int* idx = (const int*)d_in[1];
    const unsigned char* mask = (const unsigned char*)d_in[2];  // jnp.bool_ = 1 byte
    const float* W1  = (const float*)d_in[3];
    const float* b1  = (const float*)d_in[4];
    const float* Wr1 = (const float*)d_in[5];
    const float* br1 = (const float*)d_in[6];
    const float* Wr2 = (const float*)d_in[7];
    const float* br2 = (const float*)d_in[8];
    const float* W2  = (const float*)d_in[9];
    const float* b2  = (const float*)d_in[10];
    const float* W3  = (const float*)d_in[11];
    const float* b3  = (const float*)d_in[12];
    float* out = (float*)d_out;

    float* xbuf  = (float*)d_ws;                 // [N,16] running activation
    float* hbuf  = xbuf + (size_t)NVOX * CH;     // [N,16] temp / conv2
    float* c1buf = hbuf + (size_t)NVOX * CH;     // [N,16] conv1 skip

    dim3 blk(256);
    conv_in_kernel<<<NVOX / 256, blk, 0, stream>>>(feats, idx, mask, W1, b1,
                                                   c1buf, xbuf);
    for (int i = 0; i < NLAYER; ++i) {
        // h = leaky_relu(conv(x))
        conv16_wmma_kernel<<<NVOX / 256, blk, 0, stream>>>(
            xbuf, idx, mask, Wr1 + (size_t)i * KOFF * CH * CH,
            br1 + (size_t)i * CH, nullptr, hbuf, 1);
        // x = conv(h) + x   (gathers h; residual read/write of x is same-thread)
        conv16_wmma_kernel<<<NVOX / 256, blk, 0, stream>>>(
            hbuf, idx, mask, Wr2 + (size_t)i * KOFF * CH * CH,
            br2 + (size_t)i * CH, xbuf, xbuf, 0);
    }
    // conv2 = conv(x) + conv1   (reuse hbuf)
    conv16_wmma_kernel<<<NVOX / 256, blk, 0, stream>>>(
        xbuf, idx, mask, W2, b2, c1buf, hbuf, 0);
    // conv3 = conv(conv2) + feats
    conv_out_kernel<<<NVOX / 256, blk, 0, stream>>>(hbuf, idx, mask, W3, b3,
                                                    feats, out);
}